// LlamaBlock_18760417149196
// MI455X (gfx1250) — compile-verified
//
#include <hip/hip_runtime.h>
#include <math.h>

// ---------------------------------------------------------------------------
// Llama decoder block, single token, fp32, MI455X (gfx1250).
// Memory-bound (~0.83 GB streamed per call @ 23.3 TB/s => ~36us floor), so all
// GEMVs are bandwidth-shaped fp32 streams with split-K + deterministic
// reduction. Attention scores use native-fp32 V_WMMA_F32_16X16X4_F32 (no
// precision change vs reference). GEMV loop issues global_prefetch_b8.
// ---------------------------------------------------------------------------

#define HID 4096
#define NH 32
#define HD 128
#define IM 11008
#define MS 4096
#define EPSF 1e-6f

#define GTN 512   // outputs per gemv block (128 thr x float4)
#define GKC 256   // k-chunk per gemv block (split-K)
#define SCH 256   // seq chunk for AV partial

typedef __attribute__((ext_vector_type(2))) float v2f;
typedef __attribute__((ext_vector_type(4))) float v4f;
typedef __attribute__((ext_vector_type(8))) float v8f;

// ---------------- RMSNorm: out = x * w * rsqrt(mean(x^2)+eps) --------------
__global__ void k_rmsnorm(const float* __restrict__ x, const float* __restrict__ w,
                          float* __restrict__ out) {
  __shared__ float red[256];
  int t = threadIdx.x;
  float ssq = 0.f;
  for (int i = t; i < HID; i += 256) { float v = x[i]; ssq += v * v; }
  red[t] = ssq; __syncthreads();
  for (int off = 128; off > 0; off >>= 1) {
    if (t < off) red[t] += red[t + off];
    __syncthreads();
  }
  float scale = rsqrtf(red[0] * (1.0f / HID) + EPSF);
  for (int i = t; i < HID; i += 256) out[i] = x[i] * w[i] * scale;
}

// ---------------- split-K GEMV partial: part[ky][n] = sum_k x[k]*W[k][n] ---
__global__ void k_gemv_part(const float* __restrict__ W, const float* __restrict__ x,
                            float* __restrict__ part, int N, int K) {
  __shared__ float xs[GKC];
  int t = threadIdx.x;                       // 128 threads
  int n = blockIdx.x * GTN + t * 4;
  int k0 = blockIdx.y * GKC;
  int kend = (k0 + GKC < K) ? (k0 + GKC) : K;
  int kc = kend - k0;
  for (int i = t; i < kc; i += 128) xs[i] = x[k0 + i];
  __syncthreads();
  if (n >= N) return;
  v4f acc = {0.f, 0.f, 0.f, 0.f};
  const float* wp = W + (size_t)k0 * N + n;
  for (int k = 0; k < kc; ++k) {
    if (k + 8 < kc) __builtin_prefetch(wp + (size_t)8 * N, 0, 0); // global_prefetch_b8
    v4f wv = *(const v4f*)wp;                 // coalesced b128 row stream
    acc += wv * xs[k];
    wp += N;
  }
  *(v4f*)(part + (size_t)blockIdx.y * N + n) = acc;
}

// ---------------- deterministic split-K reduce + epilogue ------------------
// mode 0: raw   mode 1: resid[i] + s   mode 2: silu(s)
__global__ void k_reduce(const float* __restrict__ part, float* __restrict__ out,
                         int N, int C, int mode, const float* __restrict__ resid) {
  int i = blockIdx.x * 256 + threadIdx.x;
  if (i >= N) return;
  float s = 0.f;
  for (int c = 0; c < C; ++c) s += part[(size_t)c * N + i];
  if (mode == 1) s += resid[i];
  else if (mode == 2) s = s / (1.f + expf(-s));
  out[i] = s;
}

// ---------------- RoPE (in place, block == one head so sync is enough) -----
__global__ void k_rope(float* __restrict__ q, float* __restrict__ kv,
                       const float* __restrict__ sinc, const float* __restrict__ cosc,
                       const int* __restrict__ posp) {
  int pos = posp[0];
  int b = blockIdx.x;        // 0..31 => q heads, 32..63 => k heads
  int d = threadIdx.x;       // 0..127
  float* vec = (b < NH) ? (q + b * HD) : (kv + (b - NH) * HD);
  float v0 = vec[d];
  float vp = (d < 64) ? vec[d + 64] : vec[d - 64];
  __syncthreads();
  float rot = (d < 64) ? -vp : vp;
  vec[d] = v0 * cosc[pos * HD + d] + rot * sinc[pos * HD + d];
}

// ---------------- attention scores via V_WMMA_F32_16X16X4_F32 --------------
// One wave = (head, 16 seq positions). A = 16 cache rows x 4 dims, B = q-chunk
// broadcast along N. 32 unrolled fp32 WMMAs accumulate the 128-dim dot.
// D column N=0 (lanes 0 and 16, VGPRs 0..7) holds the 16 scores.
__global__ void k_scores(const float* __restrict__ q, const float* __restrict__ knew,
                         const float* __restrict__ kcache, const int* __restrict__ posp,
                         float* __restrict__ sc) {
  int pos = posp[0];
  int head = blockIdx.y;
  int s0 = blockIdx.x * 16;
  if (s0 > pos) return;                 // wave-uniform: EXEC stays full for WMMA
  int lane = threadIdx.x;               // 32 threads = 1 wave
  int half = lane >> 4;                 // lanes 0-15: K={0,1}; 16-31: K={2,3}
  int m = lane & 15;
  int s = s0 + m;
  const float* krow = (s == pos) ? (knew + head * HD)
                                 : (kcache + ((size_t)s * NH + head) * HD);
  const float* qrow = q + head * HD;
  v8f acc = {};
#pragma unroll
  for (int d0 = 0; d0 < HD; d0 += 4) {
    v2f a = *(const v2f*)(krow + d0 + 2 * half);   // A: 16x4 fp32 tile
    v2f b = *(const v2f*)(qrow + d0 + 2 * half);   // B: q broadcast along N
    acc = __builtin_amdgcn_wmma_f32_16x16x4_f32(false, a, false, b,
                                                (short)0, acc, false, false);
  }
  const float scale = 0.08838834764831845f;        // 1/sqrt(128)
  if (m == 0) {                                    // lane 0: M=0..7, lane 16: M=8..15
    int base = head * MS + s0 + half * 8;
#pragma unroll
    for (int j = 0; j < 8; ++j) sc[base + j] = acc[j] * scale;
  }
}

// ---------------- per-head softmax over s in [0, pos] ----------------------
__global__ void k_softmax(float* __restrict__ sc, const int* __restrict__ posp) {
  __shared__ float red[256];
  int pos = posp[0];
  int n = pos + 1;
  float* row = sc + blockIdx.x * MS;
  int t = threadIdx.x;
  float mx = -3.4e38f;
  for (int s = t; s < n; s += 256) mx = fmaxf(mx, row[s]);
  red[t] = mx; __syncthreads();
  for (int off = 128; off > 0; off >>= 1) {
    if (t < off) red[t] = fmaxf(red[t], red[t + off]);
    __syncthreads();
  }
  mx = red[0]; __syncthreads();
  float sum = 0.f;
  for (int s = t; s < n; s += 256) { float e = expf(row[s] - mx); row[s] = e; sum += e; }
  red[t] = sum; __syncthreads();
  for (int off = 128; off > 0; off >>= 1) {
    if (t < off) red[t] += red[t + off];
    __syncthreads();
  }
  float inv = 1.f / red[0];
  for (int s = t; s < n; s += 256) row[s] *= inv;
}

// ---------------- o = attn @ V, split over seq chunks ----------------------
__global__ void k_av_part(const float* __restrict__ attn, const float* __restrict__ vnew,
                          const float* __restrict__ vcache, const int* __restrict__ posp,
                          float* __restrict__ opart) {
  int pos = posp[0];
  int head = blockIdx.x;
  int c = blockIdx.y;
  int d = threadIdx.x;                   // 128 threads: contiguous 512B row loads
  int sb = c * SCH;
  int se = sb + SCH; se = (se < pos + 1) ? se : (pos + 1);
  float acc = 0.f;
  for (int s = sb; s < se; ++s) {
    float p = attn[head * MS + s];
    const float* vrow = (s == pos) ? (vnew + head * HD)
                                   : (vcache + ((size_t)s * NH + head) * HD);
    acc += p * vrow[d];
  }
  opart[(size_t)c * HID + head * HD + d] = acc;
}

__global__ void k_av_reduce(const float* __restrict__ opart, float* __restrict__ o) {
  int i = blockIdx.x * 256 + threadIdx.x;
  if (i >= HID) return;
  float s = 0.f;
  for (int c = 0; c < MS / SCH; ++c) s += opart[(size_t)c * HID + i];
  o[i] = s;
}

// ---------------------------------------------------------------------------
extern "C" void kernel_launch(void* const* d_in, const int* in_sizes, int n_in,
                              void* d_out, int out_size, void* d_ws, size_t ws_size,
                              hipStream_t stream) {
  const float* x      = (const float*)d_in[0];
  const float* w_q    = (const float*)d_in[1];
  const float* w_k    = (const float*)d_in[2];
  const float* w_v    = (const float*)d_in[3];
  const float* w_o    = (const float*)d_in[4];
  const float* anorm  = (const float*)d_in[5];
  const float* w_ff1  = (const float*)d_in[6];
  const float* w_ff2  = (const float*)d_in[7];
  const float* fnorm  = (const float*)d_in[8];
  const float* sinc   = (const float*)d_in[9];
  const float* cosc   = (const float*)d_in[10];
  const float* kcache = (const float*)d_in[11];
  const float* vcache = (const float*)d_in[12];
  const int*   posp   = (const int*)d_in[13];
  float* out = (float*)d_out;

  // workspace layout (floats); total ~414K floats (~1.65 MB)
  float* ws = (float*)d_ws;
  float* h   = ws;             // 4096
  float* q   = ws + 4096;      // 4096
  float* kn  = ws + 8192;      // 4096
  float* vn  = ws + 12288;     // 4096
  float* x1  = ws + 16384;     // 4096
  float* h2  = ws + 20480;     // 4096
  float* ff  = ws + 24576;     // 11008
  float* o   = ws + 35840;     // 4096
  float* sc  = ws + 40960;     // 32*4096
  float* op  = ws + 172032;    // 16*4096
  float* gp  = ws + 237568;    // up to 43*4096 (= 16*11008)

  dim3 gHH((HID + GTN - 1) / GTN, (HID + GKC - 1) / GKC);  // 8 x 16
  dim3 gIH((IM  + GTN - 1) / GTN, (HID + GKC - 1) / GKC);  // 22 x 16
  dim3 gHI((HID + GTN - 1) / GTN, (IM  + GKC - 1) / GKC);  // 8 x 43
  int cHH = (HID + GKC - 1) / GKC;                          // 16
  int cHI = (IM  + GKC - 1) / GKC;                          // 43

  // attn pre-norm
  k_rmsnorm<<<1, 256, 0, stream>>>(x, anorm, h);
  // QKV projections
  k_gemv_part<<<gHH, 128, 0, stream>>>(w_q, h, gp, HID, HID);
  k_reduce<<<(HID + 255) / 256, 256, 0, stream>>>(gp, q, HID, cHH, 0, nullptr);
  k_gemv_part<<<gHH, 128, 0, stream>>>(w_k, h, gp, HID, HID);
  k_reduce<<<(HID + 255) / 256, 256, 0, stream>>>(gp, kn, HID, cHH, 0, nullptr);
  k_gemv_part<<<gHH, 128, 0, stream>>>(w_v, h, gp, HID, HID);
  k_reduce<<<(HID + 255) / 256, 256, 0, stream>>>(gp, vn, HID, cHH, 0, nullptr);
  // RoPE on q and new k
  k_rope<<<2 * NH, HD, 0, stream>>>(q, kn, sinc, cosc, posp);
  // attention
  k_scores<<<dim3(MS / 16, NH), 32, 0, stream>>>(q, kn, kcache, posp, sc);
  k_softmax<<<NH, 256, 0, stream>>>(sc, posp);
  k_av_part<<<dim3(NH, MS / SCH), HD, 0, stream>>>(sc, vn, vcache, posp, op);
  k_av_reduce<<<(HID + 255) / 256, 256, 0, stream>>>(op, o);
  // o projection + residual
  k_gemv_part<<<gHH, 128, 0, stream>>>(w_o, o, gp, HID, HID);
  k_reduce<<<(HID + 255) / 256, 256, 0, stream>>>(gp, x1, HID, cHH, 1, x);
  // FFN
  k_rmsnorm<<<1, 256, 0, stream>>>(x1, fnorm, h2);
  k_gemv_part<<<gIH, 128, 0, stream>>>(w_ff1, h2, gp, IM, HID);
  k_reduce<<<(IM + 255) / 256, 256, 0, stream>>>(gp, ff, IM, cHH, 2, nullptr);
  k_gemv_part<<<gHI, 128, 0, stream>>>(w_ff2, ff, gp, HID, IM);
  k_reduce<<<(HID + 255) / 256, 256, 0, stream>>>(gp, out, HID, cHI, 1, x1);
}